// PointTransformer_76596446757373
// MI455X (gfx1250) — compile-verified
//
#include <hip/hip_runtime.h>
#include <math.h>

#define BB 4
#define NN 4096
#define KK 16
#define DIN 64
#define DT 128

typedef float v2f __attribute__((ext_vector_type(2)));
typedef float v4f __attribute__((ext_vector_type(4)));
typedef float v8f __attribute__((ext_vector_type(8)));
typedef int   v4i __attribute__((ext_vector_type(4)));

#if defined(__HIP_DEVICE_COMPILE__) && defined(__gfx1250__) && \
    __has_builtin(__builtin_amdgcn_global_load_async_to_lds_b128) && \
    __has_builtin(__builtin_amdgcn_s_wait_asynccnt)
#define PT_ASYNC_LDS 1
#endif

// ---- WMMA f32 16x16x4 fragment helpers ---------------------------------------
// A (16x4): lane l: m=l&15, g=l>>4 ; VGPR j = A[m][4s + 2g + j] (contiguous pair)
__device__ __forceinline__ v2f ldA(const float* A, int lda, int s, int lane) {
    int m = lane & 15, g = lane >> 4;
    const float* p = A + m * lda + 4 * s + 2 * g;
    v2f a; a.x = p[0]; a.y = p[1];
    return a;
}
// B (4x16): lane l: n=l&15, g=l>>4 ; VGPR j = B[4s + 2g + j][n]
__device__ __forceinline__ v2f ldB(const float* W, int ldn, int s, int lane) {
    int n = lane & 15, g = lane >> 4;
    const float* p = W + (4 * s + 2 * g) * ldn + n;
    v2f b; b.x = p[0]; b.y = p[ldn];
    return b;
}
// C/D: VGPR r holds (m = 8*(lane>>4) + r, n = lane&15)
__device__ __forceinline__ v8f bias_acc(const float* bias, int f0, int lane) {
    float bv = bias ? bias[f0 + (lane & 15)] : 0.0f;
    v8f c;
#pragma unroll
    for (int r = 0; r < 8; ++r) c[r] = bv;
    return c;
}
// Dual-accumulator GEMM step: one B fragment feeds two A streams (halves the
// global-load issue per WMMA -- weight fragments are the issue bottleneck).
__device__ __forceinline__ void gemm_tile2(const float* A0, const float* A1,
                                           int lda, const float* W, int ldn,
                                           int ks, v8f& c0, v8f& c1, int lane) {
    for (int s = 0; s < ks; ++s) {
        v2f b = ldB(W, ldn, s, lane);
        v2f a0 = ldA(A0, lda, s, lane);
        c0 = __builtin_amdgcn_wmma_f32_16x16x4_f32(false, a0, false, b, (short)0,
                                                   c0, false, false);
        v2f a1 = ldA(A1, lda, s, lane);
        c1 = __builtin_amdgcn_wmma_f32_16x16x4_f32(false, a1, false, b, (short)0,
                                                   c1, false, false);
    }
}
__device__ __forceinline__ void st_tile_lds(float* D, int ldd, int f0, v8f c,
                                            int lane, bool relu) {
    int n = lane & 15, mb = (lane >> 4) * 8;
#pragma unroll
    for (int r = 0; r < 8; ++r) {
        float v = c[r];
        if (relu) v = v > 0.0f ? v : 0.0f;
        D[(mb + r) * ldd + f0 + n] = v;
    }
}
// Contiguous global->LDS tile staging; async-DMA path on gfx1250 toolchains
// that expose the builtin (param type per hipcc diagnostic: int4 AS1 / AS3
// pointers), b128 vector copy otherwise. nfloats % 128 == 0.
__device__ __forceinline__ void stage_tile(float* lds_dst, const float* gsrc,
                                           int nfloats, int lane) {
#ifdef PT_ASYNC_LDS
    for (int t = lane * 4; t < nfloats; t += 32 * 4)
        __builtin_amdgcn_global_load_async_to_lds_b128(
            (__attribute__((address_space(1))) v4i*)(v4i*)(float*)(gsrc + t),
            (__attribute__((address_space(3))) v4i*)(v4i*)(lds_dst + t), 0, 0);
    __builtin_amdgcn_s_wait_asynccnt(0);
#else
    for (int t = lane * 4; t < nfloats; t += 32 * 4)
        *(v4f*)(lds_dst + t) = *(const v4f*)(gsrc + t);
#endif
}

// ---- K1: KNN (top-16 by squared distance, stable tie-break) ------------------
__global__ void pt_knn_kernel(const float* __restrict__ pos, int* __restrict__ knn) {
    __shared__ float tile[256 * 3];
    int tid = threadIdx.x;
    int b = blockIdx.x >> 4;
    int n = ((blockIdx.x & 15) << 8) + tid;
    const float* pb = pos + (size_t)b * NN * 3;
    float px = pb[n * 3 + 0], py = pb[n * 3 + 1], pz = pb[n * 3 + 2];
    float sqn = px * px + py * py + pz * pz;
    float td[KK];
    int ti[KK];
#pragma unroll
    for (int i = 0; i < KK; ++i) { td[i] = INFINITY; ti[i] = 0; }
    for (int t0 = 0; t0 < NN; t0 += 256) {
        for (int i = tid; i < 256 * 3; i += 256) tile[i] = pb[(size_t)t0 * 3 + i];
        __syncthreads();
        for (int j = 0; j < 256; ++j) {
            float jx = tile[j * 3], jy = tile[j * 3 + 1], jz = tile[j * 3 + 2];
            float sqj = jx * jx + jy * jy + jz * jz;
            float d = sqn + sqj - 2.0f * (px * jx + py * jy + pz * jz);
            if (d < td[KK - 1]) {          // strict: equal keeps earlier index
                int i = KK - 1;
                while (i > 0 && td[i - 1] > d) {
                    td[i] = td[i - 1]; ti[i] = ti[i - 1]; --i;
                }
                td[i] = d; ti[i] = t0 + j;
            }
        }
        __syncthreads();
    }
    int* o = knn + (size_t)(b * NN + n) * KK;
#pragma unroll
    for (int i = 0; i < KK; ++i) o[i] = ti[i];
}

// ---- K2: h = x@fc1 + b ; q/k/v = h@{wq,wk,wv}  (32 rows per wave) -----------
__global__ void pt_proj_kernel(const float* __restrict__ x,
                               const float* __restrict__ fc1_w,
                               const float* __restrict__ fc1_b,
                               const float* __restrict__ wq,
                               const float* __restrict__ wk,
                               const float* __restrict__ wv,
                               float* __restrict__ q, float* __restrict__ kf,
                               float* __restrict__ vf) {
    extern __shared__ float smem[];
    int lane = threadIdx.x & 31;
    int wib = threadIdx.x >> 5;
    float* xbuf = smem + wib * (32 * DIN + 32 * DT);
    float* hbuf = xbuf + 32 * DIN;
    int w = blockIdx.x * (blockDim.x >> 5) + wib;
    int row0 = w * 32;
    stage_tile(xbuf, x + (size_t)row0 * DIN, 32 * DIN, lane);
    for (int ft = 0; ft < 8; ++ft) {        // h: K=64 -> 16 WMMA steps x 2 tiles
        int f0 = ft * 16;
        v8f c0 = bias_acc(fc1_b, f0, lane);
        v8f c1 = c0;
        gemm_tile2(xbuf, xbuf + 16 * DIN, DIN, fc1_w + f0, DT, DIN / 4, c0, c1,
                   lane);
        st_tile_lds(hbuf, DT, f0, c0, lane, false);
        st_tile_lds(hbuf + 16 * DT, DT, f0, c1, lane, false);
    }
    const float* Ws[3] = {wq, wk, wv};
    float* Os[3] = {q, kf, vf};
    for (int m = 0; m < 3; ++m) {
        for (int ft = 0; ft < 8; ++ft) {    // K=128 -> 32 WMMA steps x 2 tiles
            int f0 = ft * 16;
            v8f c0 = bias_acc(nullptr, f0, lane);
            v8f c1 = c0;
            gemm_tile2(hbuf, hbuf + 16 * DT, DT, Ws[m] + f0, DT, DT / 4, c0, c1,
                       lane);
            int nl = lane & 15, mb = (lane >> 4) * 8;
#pragma unroll
            for (int r = 0; r < 8; ++r) {
                Os[m][(size_t)(row0 + mb + r) * DT + f0 + nl] = c0[r];
                Os[m][(size_t)(row0 + 16 + mb + r) * DT + f0 + nl] = c1[r];
            }
        }
    }
}

// ---- K3 helper: register softmax over the 16 rows of a D tile ---------------
__device__ __forceinline__ void softmax_store(v8f c, int p, int b,
                                              const int* nbr, const float* P,
                                              int f0, const float* vf,
                                              float* attn_out, float* res,
                                              int lane) {
    const float invs = 0.08838834764831845f;  // 1/sqrt(128)
    int n = lane & 15, mb = (lane >> 4) * 8;
    float m8 = c[0];
#pragma unroll
    for (int r = 1; r < 8; ++r) m8 = fmaxf(m8, c[r]);
    float mx = fmaxf(m8, __shfl_xor(m8, 16, 32));  // rows 0-7 <-> 8-15
    float e[8];
    float s8 = 0.0f;
#pragma unroll
    for (int r = 0; r < 8; ++r) { e[r] = expf((c[r] - mx) * invs); s8 += e[r]; }
    float rs = 1.0f / (s8 + __shfl_xor(s8, 16, 32));
    float acc = 0.0f;
    float* ao = attn_out + (size_t)p * KK * DT;
#pragma unroll
    for (int r = 0; r < 8; ++r) {
        int m = mb + r;
        float a = e[r] * rs;
        ao[m * DT + f0 + n] = a;
        acc += a * (vf[(size_t)(b * NN + nbr[m]) * DT + f0 + n] +
                    P[m * DT + f0 + n]);
    }
    acc += __shfl_xor(acc, 16, 32);
    if (lane < 16) res[(size_t)p * DT + f0 + lane] = acc;
}

// ---- K3: per-point attention, 2 points per wave ------------------------------
__global__ void pt_attn_kernel(const float* __restrict__ pos,
                               const int* __restrict__ knn,
                               const float* __restrict__ q,
                               const float* __restrict__ kf,
                               const float* __restrict__ vf,
                               const float* __restrict__ d1_w,
                               const float* __restrict__ d1_b,
                               const float* __restrict__ d2_w,
                               const float* __restrict__ d2_b,
                               const float* __restrict__ g1_w,
                               const float* __restrict__ g1_b,
                               const float* __restrict__ g2_w,
                               const float* __restrict__ g2_b,
                               float* __restrict__ attn_out,
                               float* __restrict__ res) {
    extern __shared__ float smem[];
    int lane = threadIdx.x & 31;
    float* A0 = smem;                 // hidden -> a_in   (point 0)
    float* A1 = A0 + KK * DT;         //                  (point 1)
    float* P0 = A1 + KK * DT;         // pos_enc
    float* P1 = P0 + KK * DT;
    float* T0 = P1 + KK * DT;         // relu(g1)
    float* T1 = T0 + KK * DT;
    float* relb = T1 + KK * DT;       // 2 x 16 x 4
    int* nbr = (int*)(relb + 128);    // 2 x 16
    int p0 = blockIdx.x * 2;
    int b0 = p0 >> 12;
    int b1 = (p0 + 1) >> 12;
    // warm L2/WGP$ for the weights (global_prefetch_b8)
    __builtin_prefetch(d2_w + lane * 32, 0, 0);
    __builtin_prefetch(g1_w + lane * 32, 0, 0);
    __builtin_prefetch(g2_w + lane * 32, 0, 0);
    if (lane < KK) {
#pragma unroll
        for (int i = 0; i < 2; ++i) {
            int p = p0 + i, b = p >> 12, n = p & (NN - 1);
            const float* pb = pos + (size_t)b * NN * 3;
            int j = knn[(size_t)p * KK + lane];
            nbr[i * KK + lane] = j;
            relb[i * 64 + lane * 4 + 0] = pb[n * 3 + 0] - pb[j * 3 + 0];
            relb[i * 64 + lane * 4 + 1] = pb[n * 3 + 1] - pb[j * 3 + 1];
            relb[i * 64 + lane * 4 + 2] = pb[n * 3 + 2] - pb[j * 3 + 2];
        }
    }
    // hidden = relu(rel @ d1 + b)  (3->128, VALU; d1_w read once for 2 points)
    for (int k = 0; k < KK; ++k) {
        float r00 = relb[k * 4], r01 = relb[k * 4 + 1], r02 = relb[k * 4 + 2];
        float r10 = relb[64 + k * 4], r11 = relb[64 + k * 4 + 1],
              r12 = relb[64 + k * 4 + 2];
        for (int f = lane; f < DT; f += 32) {
            float w0 = d1_w[f], w1 = d1_w[DT + f], w2 = d1_w[2 * DT + f];
            float bb = d1_b[f];
            float h0 = r00 * w0 + r01 * w1 + r02 * w2 + bb;
            float h1 = r10 * w0 + r11 * w1 + r12 * w2 + bb;
            A0[k * DT + f] = h0 > 0.0f ? h0 : 0.0f;
            A1[k * DT + f] = h1 > 0.0f ? h1 : 0.0f;
        }
    }
    // pos_enc = hidden @ d2 + b
    for (int ft = 0; ft < 8; ++ft) {
        int f0 = ft * 16;
        v8f c0 = bias_acc(d2_b, f0, lane);
        v8f c1 = c0;
        gemm_tile2(A0, A1, DT, d2_w + f0, DT, DT / 4, c0, c1, lane);
        st_tile_lds(P0, DT, f0, c0, lane, false);
        st_tile_lds(P1, DT, f0, c1, lane, false);
    }
    // a_in = q - k_gather + pos_enc  (overwrites A0/A1)
    const float* q0 = q + (size_t)p0 * DT;
    const float* q1 = q + (size_t)(p0 + 1) * DT;
    for (int t = lane; t < KK * DT; t += 32) {
        int k = t >> 7, f = t & (DT - 1);
        A0[t] = q0[f] - kf[(size_t)(b0 * NN + nbr[k]) * DT + f] + P0[t];
        A1[t] = q1[f] - kf[(size_t)(b1 * NN + nbr[KK + k]) * DT + f] + P1[t];
    }
    // t = relu(a_in @ g1 + b)
    for (int ft = 0; ft < 8; ++ft) {
        int f0 = ft * 16;
        v8f c0 = bias_acc(g1_b, f0, lane);
        v8f c1 = c0;
        gemm_tile2(A0, A1, DT, g1_w + f0, DT, DT / 4, c0, c1, lane);
        st_tile_lds(T0, DT, f0, c0, lane, true);
        st_tile_lds(T1, DT, f0, c1, lane, true);
    }
    // logits = t @ g2 + b ; softmax over K entirely in the D-tile registers
    for (int ft = 0; ft < 8; ++ft) {
        int f0 = ft * 16;
        v8f c0 = bias_acc(g2_b, f0, lane);
        v8f c1 = c0;
        gemm_tile2(T0, T1, DT, g2_w + f0, DT, DT / 4, c0, c1, lane);
        softmax_store(c0, p0, b0, nbr, P0, f0, vf, attn_out, res, lane);
        softmax_store(c1, p0 + 1, b1, nbr + KK, P1, f0, vf, attn_out, res, lane);
    }
}

// ---- K4: out = res @ fc2 + b + x  (32 rows per wave) -------------------------
__global__ void pt_out_kernel(const float* __restrict__ res,
                              const float* __restrict__ fc2_w,
                              const float* __restrict__ fc2_b,
                              const float* __restrict__ x,
                              float* __restrict__ out) {
    extern __shared__ float smem[];
    int lane = threadIdx.x & 31;
    int wib = threadIdx.x >> 5;
    float* rbuf = smem + wib * (32 * DT);
    int w = blockIdx.x * (blockDim.x >> 5) + wib;
    int row0 = w * 32;
    stage_tile(rbuf, res + (size_t)row0 * DT, 32 * DT, lane);
    for (int ft = 0; ft < 4; ++ft) {
        int f0 = ft * 16;
        v8f c0 = bias_acc(fc2_b, f0, lane);
        v8f c1 = c0;
        gemm_tile2(rbuf, rbuf + 16 * DT, DT, fc2_w + f0, DIN, DT / 4, c0, c1,
                   lane);
        int nl = lane & 15, mb = (lane >> 4) * 8;
#pragma unroll
        for (int r = 0; r < 8; ++r) {
            size_t o0 = (size_t)(row0 + mb + r) * DIN + f0 + nl;
            size_t o1 = (size_t)(row0 + 16 + mb + r) * DIN + f0 + nl;
            out[o0] = c0[r] + x[o0];
            out[o1] = c1[r] + x[o1];
        }
    }
}

extern "C" void kernel_launch(void* const* d_in, const int* in_sizes, int n_in,
                              void* d_out, int out_size, void* d_ws, size_t ws_size,
                              hipStream_t stream) {
    const float* x     = (const float*)d_in[0];
    const float* pos   = (const float*)d_in[1];
    const float* fc1_w = (const float*)d_in[2];
    const float* fc1_b = (const float*)d_in[3];
    const float* fc2_w = (const float*)d_in[4];
    const float* fc2_b = (const float*)d_in[5];
    const float* d1_w  = (const float*)d_in[6];
    const float* d1_b  = (const float*)d_in[7];
    const float* d2_w  = (const float*)d_in[8];
    const float* d2_b  = (const float*)d_in[9];
    const float* g1_w  = (const float*)d_in[10];
    const float* g1_b  = (const float*)d_in[11];
    const float* g2_w  = (const float*)d_in[12];
    const float* g2_b  = (const float*)d_in[13];
    const float* wq    = (const float*)d_in[14];
    const float* wk    = (const float*)d_in[15];
    const float* wv    = (const float*)d_in[16];

    float* out_y    = (float*)d_out;                      // [B,N,64]
    float* out_attn = out_y + (size_t)BB * NN * DIN;      // [B,N,K,128]

    char* ws = (char*)d_ws;
    int*   knn = (int*)ws;   ws += (size_t)BB * NN * KK * sizeof(int);
    float* q   = (float*)ws; ws += (size_t)BB * NN * DT * sizeof(float);
    float* kf  = (float*)ws; ws += (size_t)BB * NN * DT * sizeof(float);
    float* vf  = (float*)ws; ws += (size_t)BB * NN * DT * sizeof(float);
    float* res = (float*)ws;

    pt_knn_kernel<<<64, 256, 0, stream>>>(pos, knn);
    // 512 waves x 32 rows, 2 waves/block, 48KB LDS
    pt_proj_kernel<<<256, 64, 2 * (32 * DIN + 32 * DT) * sizeof(float), stream>>>(
        x, fc1_w, fc1_b, wq, wk, wv, q, kf, vf);
    // 8192 single-wave blocks, 2 points each, ~48.6KB LDS
    pt_attn_kernel<<<8192, 32, (6 * KK * DT + 128 + 32) * sizeof(float), stream>>>(
        pos, knn, q, kf, vf, d1_w, d1_b, d2_w, d2_b, g1_w, g1_b, g2_w, g2_b,
        out_attn, res);
    // 512 waves x 32 rows, 2 waves/block, 32KB LDS
    pt_out_kernel<<<256, 64, 2 * (32 * DT) * sizeof(float), stream>>>(
        res, fc2_w, fc2_b, x, out_y);
}